// GELU5_45844480918256
// MI455X (gfx1250) — compile-verified
//
#include <hip/hip_runtime.h>
#include <hip/hip_bf16.h>

// Problem constants from the reference: B=16, T=2048, D=1024, K=8.
#define GD   1024
#define GK   8
#define GBT  32768           // B*T rows
#define MAIN_TPB 256         // one row (1024 floats) = 256 threads * float4
#define RPB  16              // rows per block
#define DEPTH 4              // async LDS pipeline depth

typedef __attribute__((ext_vector_type(2))) float v2f;
typedef __attribute__((ext_vector_type(4))) float f4;
typedef __attribute__((ext_vector_type(8))) float v8f;

#if __has_builtin(__builtin_amdgcn_wmma_f32_16x16x4_f32)
#define HAVE_WMMA4 1
#else
#define HAVE_WMMA4 0
#endif

// ---- CDNA5 async global->LDS copy (ASYNCcnt-tracked) -----------------------
__device__ __forceinline__ void async_load_f4(unsigned lds_off, const f4* g) {
  // VDST = per-lane LDS byte offset, VADDR = per-lane 64-bit global address.
  asm volatile("global_load_async_to_lds_b128 %0, %1, off"
               :: "v"(lds_off), "v"(g)
               : "memory");
}

// ---- stage 0: zero the column-sum accumulator ------------------------------
__global__ __launch_bounds__(256) void zero_ws(float* __restrict__ colsum) {
  for (int d = threadIdx.x; d < GD; d += 256) colsum[d] = 0.0f;
}

// ---- stage 1: column sums of x (for x_mean) --------------------------------
__global__ __launch_bounds__(256) void colsum_kernel(const float* __restrict__ x,
                                                     float* __restrict__ colsum) {
  const int tid = threadIdx.x;
  const long long row0 = (long long)blockIdx.x * RPB;
  const f4* xg = (const f4*)x + row0 * (GD / 4) + tid;
  f4 acc = {0.0f, 0.0f, 0.0f, 0.0f};
#pragma unroll
  for (int i = 0; i < RPB; ++i) {
    f4 v = xg[i * (GD / 4)];
    acc.x += v.x; acc.y += v.y; acc.z += v.z; acc.w += v.w;
  }
  float* c = colsum + tid * 4;
  atomicAdd(c + 0, acc.x);
  atomicAdd(c + 1, acc.y);
  atomicAdd(c + 2, acc.z);
  atomicAdd(c + 3, acc.w);
}

// ---- stage 2: distances (WMMA Gram diagonal), softmax r, P/Q/tau/alpha -----
__global__ __launch_bounds__(256) void prep_kernel(
    const float* __restrict__ colsum, const float* __restrict__ mus,
    const float* __restrict__ wsK, const float* __restrict__ log_tau,
    const float* __restrict__ log_blend,
    float* __restrict__ tauB, float* __restrict__ alphaB,
    float* __restrict__ Pb, float* __restrict__ Qb) {
  __shared__ float xm[GD];
  __shared__ float distsq[GK];
  __shared__ float rsh[GK];
#if HAVE_WMMA4
  __shared__ float gram[8][32][8];
#else
  __shared__ float red[256];
#endif
  const int tid = threadIdx.x;
  for (int d = tid; d < GD; d += 256) xm[d] = colsum[d] * (1.0f / GBT);
  __syncthreads();

#if HAVE_WMMA4
  {
    // D = A(16x4) x B(4x16): A holds diff rows for clusters m=0..7 (8..15 zero).
    // For the symmetric Gram (B = A^T) the A and B VGPR layouts coincide, so
    // one register image feeds both operands. dist^2_k = diagonal element k.
    const int lane = tid & 31, wave = tid >> 5;
    const int m = lane & 15;
    const int mm = (m < GK) ? m : 0;
    const float msk = (m < GK) ? 1.0f : 0.0f;
    const int dsel = (lane < 16) ? 0 : 2;   // lanes 0-15 hold K=0,1; 16-31 hold K=2,3
    v8f acc = {};
    for (int cc = 0; cc < 32; ++cc) {       // each wave covers 32 K-chunks of 4
      const int d0 = (wave * 32 + cc) * 4 + dsel;
      v2f a;
      a.x = msk * (xm[d0]     - mus[mm * GD + d0]);
      a.y = msk * (xm[d0 + 1] - mus[mm * GD + d0 + 1]);
      acc = __builtin_amdgcn_wmma_f32_16x16x4_f32(false, a, false, a,
                                                  (short)0, acc, false, false);
    }
#pragma unroll
    for (int j = 0; j < 8; ++j) gram[wave][lane][j] = acc[j];
  }
  __syncthreads();
  if (tid < GK) {
    float s = 0.0f;
#pragma unroll
    for (int w = 0; w < 8; ++w) s += gram[w][tid][tid];  // vgpr k, lane k = (k,k)
    distsq[tid] = s;
  }
  __syncthreads();
#else
  for (int k = 0; k < GK; ++k) {
    float local = 0.0f;
    for (int d = tid; d < GD; d += 256) {
      float df = xm[d] - mus[k * GD + d];
      local += df * df;
    }
    red[tid] = local; __syncthreads();
    for (int s = 128; s > 0; s >>= 1) {
      if (tid < s) red[tid] += red[tid + s];
      __syncthreads();
    }
    if (tid == 0) distsq[k] = red[0];
    __syncthreads();
  }
#endif

  if (tid == 0) {
    float lg[GK], mx = -1e30f;
#pragma unroll
    for (int k = 0; k < GK; ++k) {
      lg[k] = __logf(wsK[k]) - sqrtf(distsq[k] * (1.0f / GD));
      mx = fmaxf(mx, lg[k]);
    }
    float sum = 0.0f;
#pragma unroll
    for (int k = 0; k < GK; ++k) { lg[k] = __expf(lg[k] - mx); sum += lg[k]; }
    const float inv = 1.0f / sum;
#pragma unroll
    for (int k = 0; k < GK; ++k) rsh[k] = lg[k] * inv;
  }
  __syncthreads();

  // r_k*exp(-tau*|x-mu|) = min(e^{tau x} * Q_k, e^{-tau x} * P_k)
  for (int d = tid; d < GD; d += 256) {
    const float tau = __expf(log_tau[d]);
    const float alpha = 1.0f / (1.0f + __expf(-log_blend[d]));
    tauB[d] = tau;
    alphaB[d] = alpha;
#pragma unroll
    for (int k = 0; k < GK; ++k) {
      const float mu = mus[k * GD + d];
      Pb[k * GD + d] = rsh[k] * __expf(tau * mu);
      Qb[k * GD + d] = rsh[k] * __expf(-tau * mu);
    }
  }
}

// ---- stage 3: main streaming pass ------------------------------------------
// out = y * sigmoid(2*c*(y + 0.044715 y^3)) == 0.5*y*(1+tanh(...)), one exp+rcp.
#define GELU5_COMP(comp)                                                     \
  {                                                                          \
    const float xs = xv.comp, tau = tau4.comp, al = al4.comp;                \
    const float t = __expf(tau * xs);                                        \
    const float rt = __builtin_amdgcn_rcpf(t);                               \
    float fam = 0.0f;                                                        \
    _Pragma("unroll")                                                        \
    for (int k = 0; k < GK; ++k)                                             \
      fam += fminf(t * Q4[k].comp, rt * P4[k].comp);                         \
    const float y = xs * (1.0f - al * fam);                                  \
    const float y2 = y * y;                                                  \
    const float e = __expf(y * (-1.5957691216f - 0.0713548162f * y2));       \
    res.comp = y * __builtin_amdgcn_rcpf(1.0f + e);                          \
  }

__global__ __launch_bounds__(256) void gelu5_main(
    const float* __restrict__ x, const float* __restrict__ tauB,
    const float* __restrict__ alphaB, const float* __restrict__ Pb,
    const float* __restrict__ Qb, float* __restrict__ out) {
  __shared__ f4 buf[DEPTH][MAIN_TPB];
  const int tid = threadIdx.x;
  const int d0 = tid * 4;

  // Per-thread column constants live in registers for the whole block.
  const f4 tau4 = *(const f4*)(tauB + d0);
  const f4 al4  = *(const f4*)(alphaB + d0);
  f4 P4[GK], Q4[GK];
#pragma unroll
  for (int k = 0; k < GK; ++k) {
    P4[k] = *(const f4*)(Pb + k * GD + d0);
    Q4[k] = *(const f4*)(Qb + k * GD + d0);
  }

  const long long row0 = (long long)blockIdx.x * RPB;
  const f4* xg = (const f4*)x + row0 * (GD / 4) + tid;
  f4* og = (f4*)out + row0 * (GD / 4) + tid;
  // Low 32 bits of a flat LDS pointer == LDS byte offset (aperture in hi bits).
  const unsigned lds0 = (unsigned)(unsigned long long)&buf[0][tid];

  // Prologue: fill the 4-deep async pipeline.
#pragma unroll
  for (int p = 0; p < DEPTH; ++p)
    async_load_f4(lds0 + (unsigned)(p * (MAIN_TPB * 16)), xg + p * (GD / 4));

#pragma unroll 1
  for (int i = 0; i < RPB - DEPTH; ++i) {
    asm volatile("s_wait_asynccnt 3" ::: "memory");        // oldest row landed
    const f4 xv = buf[i & (DEPTH - 1)][tid];               // ds_load_b128
    asm volatile("s_wait_dscnt 0" ::: "memory");           // WAR guard vs async
    async_load_f4(lds0 + (unsigned)((i & (DEPTH - 1)) * (MAIN_TPB * 16)),
                  xg + (i + DEPTH) * (GD / 4));            // refill slot
    f4 res;
    GELU5_COMP(x) GELU5_COMP(y) GELU5_COMP(z) GELU5_COMP(w)
    __builtin_nontemporal_store(res, og + i * (GD / 4));   // NT: keep x in L2
  }

#define GELU5_TAIL(J, WAITSTR)                                               \
  {                                                                          \
    asm volatile(WAITSTR ::: "memory");                                      \
    const f4 xv = buf[(RPB - DEPTH + J) & (DEPTH - 1)][tid];                 \
    f4 res;                                                                  \
    GELU5_COMP(x) GELU5_COMP(y) GELU5_COMP(z) GELU5_COMP(w)                  \
    __builtin_nontemporal_store(res, og + (RPB - DEPTH + J) * (GD / 4));     \
  }
  GELU5_TAIL(0, "s_wait_asynccnt 3")
  GELU5_TAIL(1, "s_wait_asynccnt 2")
  GELU5_TAIL(2, "s_wait_asynccnt 1")
  GELU5_TAIL(3, "s_wait_asynccnt 0")
#undef GELU5_TAIL
}

// ---- host-side launch ------------------------------------------------------
extern "C" void kernel_launch(void* const* d_in, const int* in_sizes, int n_in,
                              void* d_out, int out_size, void* d_ws, size_t ws_size,
                              hipStream_t stream) {
  (void)in_sizes; (void)n_in; (void)out_size; (void)ws_size;
  const float* x      = (const float*)d_in[0];
  const float* mus    = (const float*)d_in[1];
  const float* wsK    = (const float*)d_in[2];
  const float* ltau   = (const float*)d_in[3];
  const float* lblend = (const float*)d_in[4];

  float* ws     = (float*)d_ws;
  float* colsum = ws;                  // [1024]
  float* tauB   = ws + 1024;           // [1024]
  float* alphaB = ws + 2048;           // [1024]
  float* Pb     = ws + 3072;           // [8*1024]
  float* Qb     = ws + 3072 + GK * GD; // [8*1024]
  float* out    = (float*)d_out;

  zero_ws<<<1, 256, 0, stream>>>(colsum);
  colsum_kernel<<<GBT / RPB, 256, 0, stream>>>(x, colsum);
  prep_kernel<<<1, 256, 0, stream>>>(colsum, mus, wsK, ltau, lblend,
                                     tauB, alphaB, Pb, Qb);
  gelu5_main<<<GBT / RPB, 256, 0, stream>>>(x, tauB, alphaB, Pb, Qb, out);
}